// GCNEncoder_893353197858
// MI455X (gfx1250) — compile-verified
//
#include <hip/hip_runtime.h>

typedef __attribute__((ext_vector_type(2))) float v2f;
typedef __attribute__((ext_vector_type(4))) float f4;
typedef __attribute__((ext_vector_type(8))) float v8f;

#define NNODES 100000

__device__ __forceinline__ void atomicAddF32(float* p, float v) {
    __hip_atomic_fetch_add(p, v, __ATOMIC_RELAXED, __HIP_MEMORY_SCOPE_AGENT);
}

// ---------------- degree / norm ----------------

__global__ void k_zero_u32(unsigned int* __restrict__ p, int n) {
    int i = blockIdx.x * blockDim.x + threadIdx.x;
    if (i < n) p[i] = 0u;
}

__global__ void k_deg(const long long* __restrict__ dst, unsigned int* __restrict__ deg, int E) {
    int i = blockIdx.x * blockDim.x + threadIdx.x;
    if (i < E) atomicAdd(&deg[(int)dst[i]], 1u);
}

__global__ void k_dinv(const unsigned int* __restrict__ deg, float* __restrict__ dinv, int n) {
    int i = blockIdx.x * blockDim.x + threadIdx.x;
    if (i < n) dinv[i] = rsqrtf((float)(deg[i] + 1u));  // +1 self loop => deg >= 1 always
}

// ---------------- fp32 WMMA GEMM: C[M,NC] = A[M,128] @ B[128,NC] ----------------
// Block = 256 threads (8 waves). Block covers 128 rows x NC cols.
// Wave w owns rows [16w, 16w+16) and iterates all NC/16 column tiles.
// B is staged once per block into LDS, TRANSPOSED (Bs[col][k], stride 130) so the
// per-lane K-pair fragment is a single 8B-aligned ds_load_b64.
template <int NC>
__global__ void __launch_bounds__(256) k_gemm_wmma_f32(const float* __restrict__ A,
                                                       const float* __restrict__ B,
                                                       float* __restrict__ C, int M) {
    constexpr int K   = 128;
    constexpr int NT  = NC / 16;   // column tiles per wave
    constexpr int LDB = K + 2;     // 130: keeps (col*LDB + even k) 8B-aligned, spreads banks
    __shared__ float Bs[NC * LDB];

    const int tid = threadIdx.x;
    // Stage B transposed: Bs[col*LDB + k] = B[k*NC + col]; coalesced global reads.
    for (int idx = tid; idx < K * NC; idx += 256) {
        int k   = idx / NC;
        int col = idx % NC;
        Bs[col * LDB + k] = B[idx];
    }
    __syncthreads();

    const int wave = tid >> 5;
    const int lane = tid & 31;
    const int half = lane >> 4;    // 0: lanes 0-15, 1: lanes 16-31
    const int l16  = lane & 15;
    const int row0 = blockIdx.x * 128 + wave * 16;
    if (row0 >= M) return;         // tail block: whole-wave guard (after barrier)

    const float* Arow = A + (size_t)(row0 + l16) * K;   // A row M = l16 of this tile

    v8f acc[NT];
#pragma unroll
    for (int t = 0; t < NT; ++t) acc[t] = (v8f){0.f, 0.f, 0.f, 0.f, 0.f, 0.f, 0.f, 0.f};

    for (int k0 = 0; k0 < K; k0 += 4) {
        const int kb = k0 + half * 2;
        // A 16x4 frag: VGPR0 = A[M][kb], VGPR1 = A[M][kb+1] -> one global v2f load
        v2f a = *(const v2f*)(Arow + kb);
#pragma unroll
        for (int t = 0; t < NT; ++t) {
            // B 4x16 frag: VGPR0 = B[kb][col], VGPR1 = B[kb+1][col] -> one ds_load_b64
            v2f b = *(const v2f*)(&Bs[(t * 16 + l16) * LDB + kb]);
            acc[t] = __builtin_amdgcn_wmma_f32_16x16x4_f32(
                /*neg_a=*/false, a, /*neg_b=*/false, b,
                /*c_mod=*/(short)0, acc[t], /*reuse_a=*/false, /*reuse_b=*/false);
        }
    }
    // D layout: VGPR i -> row (i + half*8), column = t*16 + l16
    float* Cp = C + (size_t)(row0 + half * 8) * NC + l16;
#pragma unroll
    for (int t = 0; t < NT; ++t)
#pragma unroll
        for (int i = 0; i < 8; ++i)
            Cp[(size_t)i * NC + t * 16] = acc[t][i];
}

// ---------------- self-loop init: acc[i,:] = h[i,:] * dinv[i]^2 ----------------
__global__ void k_self_init(const float* __restrict__ h, const float* __restrict__ dinv,
                            float* __restrict__ acc, int n, int q4shift) {
    int i = blockIdx.x * blockDim.x + threadIdx.x;   // over n * (F/4)
    int total = n << q4shift;
    if (i >= total) return;
    int node = i >> q4shift;
    float w = dinv[node];
    w *= w;
    f4 v = ((const f4*)h)[i];
    f4 r = {v.x * w, v.y * w, v.z * w, v.w * w};
    ((f4*)acc)[i] = r;
}

// ---------------- edge scatter-add: acc[dst,:] += h[src,:] * dinv[s]*dinv[d] ----------------
__global__ void k_edge_scatter(const float* __restrict__ h, const float* __restrict__ dinv,
                               const long long* __restrict__ src, const long long* __restrict__ dst,
                               float* __restrict__ acc, int E, int q4shift) {
    long long gid = (long long)blockIdx.x * blockDim.x + threadIdx.x;
    long long total = (long long)E << q4shift;
    if (gid >= total) return;
    int e = (int)(gid >> q4shift);
    int q = (int)(gid & ((1 << q4shift) - 1));
    int s = (int)src[e];
    int d = (int)dst[e];
    float w = dinv[s] * dinv[d];
    const int F = 4 << q4shift;
    f4 v = *(const f4*)(h + (size_t)s * F + 4 * q);
    float* p = acc + (size_t)d * F + 4 * q;
    atomicAddF32(p + 0, v.x * w);
    atomicAddF32(p + 1, v.y * w);
    atomicAddF32(p + 2, v.z * w);
    atomicAddF32(p + 3, v.w * w);
}

// ---------------- bias + BatchNorm(eval) + ReLU (F = 128), float4 ----------------
__global__ void k_bnrelu(const float* __restrict__ acc, const float* __restrict__ b1,
                         const float* __restrict__ gamma, const float* __restrict__ beta,
                         const float* __restrict__ rmean, const float* __restrict__ rvar,
                         float* __restrict__ out, int n) {
    int i = blockIdx.x * blockDim.x + threadIdx.x;   // over n*32
    if (i >= n * 32) return;
    int f0 = (i & 31) * 4;
    f4 v = ((const f4*)acc)[i];
    float r[4] = {v.x, v.y, v.z, v.w};
    f4 o;
#pragma unroll
    for (int j = 0; j < 4; ++j) {
        int f = f0 + j;
        float t = (r[j] + b1[f] - rmean[f]) * rsqrtf(rvar[f] + 1e-5f) * gamma[f] + beta[f];
        ((float*)&o)[j] = t > 0.f ? t : 0.f;
    }
    ((f4*)out)[i] = o;
}

// ---------------- output init: out[i,:] = g[i,:]*dinv[i]^2 + b2 (F = 64) ----------------
__global__ void k_out_init(const float* __restrict__ g, const float* __restrict__ dinv,
                           const float* __restrict__ b2, float* __restrict__ out, int n) {
    int i = blockIdx.x * blockDim.x + threadIdx.x;   // over n*16
    if (i >= n * 16) return;
    int node = i >> 4;
    int q = i & 15;
    float w = dinv[node];
    w *= w;
    f4 v = ((const f4*)g)[i];
    const float* bb = b2 + 4 * q;
    f4 r = {v.x * w + bb[0], v.y * w + bb[1], v.z * w + bb[2], v.w * w + bb[3]};
    ((f4*)out)[i] = r;
}

extern "C" void kernel_launch(void* const* d_in, const int* in_sizes, int n_in,
                              void* d_out, int out_size, void* d_ws, size_t ws_size,
                              hipStream_t stream) {
    const float*     x     = (const float*)d_in[0];
    const long long* ei    = (const long long*)d_in[1];   // int64 [2, E]
    const float*     W1    = (const float*)d_in[2];
    const float*     b1    = (const float*)d_in[3];
    const float*     gamma = (const float*)d_in[4];
    const float*     beta  = (const float*)d_in[5];
    const float*     rmean = (const float*)d_in[6];
    const float*     rvar  = (const float*)d_in[7];
    const float*     W2    = (const float*)d_in[8];
    const float*     b2    = (const float*)d_in[9];
    float* out = (float*)d_out;

    const int n = NNODES;
    const int E = in_sizes[1] / 2;
    const long long* src = ei;
    const long long* dst = ei + E;

    // Workspace layout
    char* ws = (char*)d_ws;
    size_t off = 0;
    float* dinv = (float*)(ws + off);              off += ((size_t)n * 4 + 511) & ~(size_t)511;
    unsigned int* deg = (unsigned int*)(ws + off); off += ((size_t)n * 4 + 511) & ~(size_t)511;
    float* h1   = (float*)(ws + off);              off += (size_t)n * 128 * 4;
    float* acc1 = (float*)(ws + off);              off += (size_t)n * 128 * 4;
    float* h2 = h1;    // reuse: bnrelu reads acc1, writes h1's buffer
    float* g  = acc1;  // reuse: gemm2 reads h2 (h1 buf), writes acc1's buffer

    const int B = 256;
    // 1) degrees + dinv
    k_zero_u32<<<(n + B - 1) / B, B, 0, stream>>>(deg, n);
    k_deg<<<(E + B - 1) / B, B, 0, stream>>>(dst, deg, E);
    k_dinv<<<(n + B - 1) / B, B, 0, stream>>>(deg, dinv, n);

    // 2) h1 = x @ W1   (M=100000, K=128, NC=128; 128 rows/block)
    k_gemm_wmma_f32<128><<<(n + 127) / 128, 256, 0, stream>>>(x, W1, h1, n);

    // 3) aggregate layer 1 (F=128, q4shift=5)
    k_self_init<<<((n << 5) + B - 1) / B, B, 0, stream>>>(h1, dinv, acc1, n, 5);
    {
        long long total = (long long)E << 5;
        k_edge_scatter<<<(unsigned)((total + B - 1) / B), B, 0, stream>>>(h1, dinv, src, dst, acc1, E, 5);
    }

    // 4) +b1, BN eval, ReLU -> h2
    k_bnrelu<<<((n * 32) + B - 1) / B, B, 0, stream>>>(acc1, b1, gamma, beta, rmean, rvar, h2, n);

    // 5) g = h2 @ W2   (K=128, NC=64)
    k_gemm_wmma_f32<64><<<(n + 127) / 128, 256, 0, stream>>>(h2, W2, g, n);

    // 6) aggregate layer 2 directly into d_out (F=64, q4shift=4)
    k_out_init<<<((n << 4) + B - 1) / B, B, 0, stream>>>(g, dinv, b2, out, n);
    {
        long long total = (long long)E << 4;
        k_edge_scatter<<<(unsigned)((total + B - 1) / B), B, 0, stream>>>(g, dinv, src, dst, out, E, 4);
    }
}